// tb_floquet_tbc_cuda_23811298689162
// MI455X (gfx1250) — compile-verified
//
#include <hip/hip_runtime.h>

// ---------------------------------------------------------------------------
// Problem constants (fixed by the reference: NX = NY = 64).
// Output: [4, 4096, 4096] complex128, row-major, viewed here as double pairs.
// ---------------------------------------------------------------------------
constexpr int NXc    = 64;
constexpr int NYc    = 64;
constexpr int NSITES = NXc * NYc;          // 4096
constexpr int MAXP   = 4608;               // upper bound on pairs per H (~2050 actual)
constexpr int NTW    = 32;                 // twist links per H (exact for 64x64)

struct PairTab {
  int n;                // number of (r,c) pairs
  int r[MAXP];
  int c[MAXP];
  int tr[NTW];
  int tc[NTW];
};

// ---------------------------------------------------------------------------
// Compile-time transcriptions of the Python pair generators (_h1.._h4_pairs).
// ---------------------------------------------------------------------------
constexpr PairTab mk_h1() {
  PairTab t{};
  int pa[NXc] = {};
  int pb[NXc] = {};
  pa[0] = 0; pb[0] = NXc;
  for (int j = 1; j < NXc; ++j) {
    int inc = (j == NXc / 2) ? 3 : 2;
    pa[j] = pa[j - 1] + inc;
    pb[j] = pb[j - 1] + inc;
  }
  for (int j = 0; j < NXc; ++j) {
    int a = pa[j], b = pb[j];
    while (a < NSITES && b < NSITES) {
      t.r[t.n] = a; t.c[t.n] = b; ++t.n;
      a += 2 * NXc; b += 2 * NXc;
    }
  }
  int p = 0;
  while (1 + 2 * p < NXc) {                      // ny even
    t.tr[p] = 1 + NXc * (NYc - 1) + 2 * p;
    t.tc[p] = 1 + 2 * p;
    ++p;
  }
  return t;
}

constexpr PairTab mk_h2() {
  PairTab t{};
  int n = 1, a = -1, b = 0;
  while (n <= NYc && a < NSITES - 1 && b < NSITES) {
    a += 2; b += 2;
    if (b < n * NXc) {
      t.r[t.n] = a; t.c[t.n] = b; ++t.n;
    } else {
      n += 1; a += 1; b += 1;                    // nx even
      if (a < NSITES - 1 && b < NSITES) { t.r[t.n] = a; t.c[t.n] = b; ++t.n; }
    }
    if (b >= NSITES - 2) break;
  }
  int p = 0;
  while (NXc - 1 + 2 * NXc * p < NSITES) {       // nx even
    t.tr[p] = NXc - 1 + 2 * NXc * p;
    t.tc[p] = 2 * NXc * p;
    ++p;
  }
  return t;
}

constexpr PairTab mk_h3() {
  PairTab t{};
  int n = 1, a = 1, b = 1 + NXc;
  if (b < NSITES) { t.r[t.n] = a; t.c[t.n] = b; ++t.n; }
  while (n < NYc && a < NSITES - 1 && b < NSITES - 1) {
    a += 2; b += 2;
    if (a < n * NXc) {
      t.r[t.n] = a; t.c[t.n] = b; ++t.n;
    } else {
      n += 1;
      if (n % 2 == 0)            { a -= 1; b -= 1; }
      else if (b < NSITES - 1)   { a += 1; b += 1; }   // n%2 != 0 branch
      else                       { a -= 2; b -= 2; }
      t.r[t.n] = a; t.c[t.n] = b; ++t.n;
    }
  }
  int p = 0;
  while (2 * p < NXc) {                          // ny even
    t.tr[p] = NXc * (NYc - 1) + 2 * p;
    t.tc[p] = 2 * p;
    ++p;
  }
  return t;
}

constexpr PairTab mk_h4() {
  PairTab t{};
  int n = 1, a = -2, b = -1;
  while (n <= NYc && a < NSITES - 2 && b < NSITES - 2) {
    a += 2; b += 2;
    if (b < n * NXc) {
      t.r[t.n] = a; t.c[t.n] = b; ++t.n;
    } else {
      n += 1; a += 1; b += 1;                    // nx even, nx != 2
      t.r[t.n] = a; t.c[t.n] = b; ++t.n;
    }
  }
  int p = 0;
  while (2 * NXc * (1 + p) - 1 < NSITES) {       // nx even
    t.tr[p] = 2 * NXc * (1 + p) - 1;
    t.tc[p] = 2 * NXc * p + NXc;
    ++p;
  }
  return t;
}

// Baked into device rodata at compile time (~148 KB total).
__device__ const PairTab g_tabs[4] = { mk_h1(), mk_h2(), mk_h3(), mk_h4() };

// ---------------------------------------------------------------------------
// Kernel 1: zero-fill 1.074 GB with 128-bit NONTEMPORAL stores.
// Output >> 192 MB L2, so TH=NT is the right cache policy on gfx1250.
//
// Launch shape divides the output exactly:
//   16384 blocks x 256 threads x 16 chunks x 16 B = 1,073,741,824 B.
// Each block owns a contiguous 64 KB tile; a thread's 16 stores are at byte
// offsets k*4096 from one base address (all fit the 24-bit immediate offset),
// so the unrolled body is pure back-to-back global_store_b128 th:NT with no
// per-store address math, compare, or branch. Lanes write consecutive 16-B
// chunks -> perfectly coalesced.
// ---------------------------------------------------------------------------
typedef unsigned int v4u __attribute__((ext_vector_type(4)));

constexpr int    FILL_BLOCKS  = 16384;
constexpr int    FILL_THREADS = 256;                    // 8 waves (wave32)
constexpr int    FILL_ITERS   = 16;
constexpr size_t TOTAL_CHUNKS = (size_t)4 * NSITES * NSITES;  // 16-B chunks
static_assert((size_t)FILL_BLOCKS * FILL_THREADS * FILL_ITERS == TOTAL_CHUNKS,
              "fill launch shape must tile the output exactly");

__global__ void __launch_bounds__(FILL_THREADS)
fill_zero_nt(v4u* __restrict__ p) {
  // Block-contiguous tiling: chunk = blk*(256*16) + k*256 + tid
  v4u* base = p + (size_t)blockIdx.x * (FILL_THREADS * FILL_ITERS) + threadIdx.x;
  v4u z = (v4u)(0u);
#pragma unroll
  for (int k = 0; k < FILL_ITERS; ++k) {
    __builtin_nontemporal_store(z, base + (size_t)k * FILL_THREADS);
  }
}

// ---------------------------------------------------------------------------
// Kernel 2: scatter the sparse Hamiltonian entries.
//   H[r,c] = -1+0i, H[c,r] = -1+0i
//   H[tr,tc] = -e^{i th} = (-cos, -sin);  H[tc,tr] = conj = (-cos, +sin)
// Twist angle: H1,H3 -> theta_y ; H2,H4 -> theta_x.
// Pair offsets (+-1 / +-64) never coincide with twist offsets (+-63 / +-4032),
// so pair and twist writes are disjoint -> one parallel kernel is race-free.
// OOB indices dropped (JAX .at[].set semantics).
// ---------------------------------------------------------------------------
typedef double v2d __attribute__((ext_vector_type(2)));

__device__ __forceinline__ void put_entry(double* __restrict__ out, int h,
                                          int r, int c, double re, double im) {
  if ((unsigned)r < (unsigned)NSITES && (unsigned)c < (unsigned)NSITES) {
    size_t idx = (((size_t)h * NSITES + (size_t)r) * NSITES + (size_t)c) * 2u;
    v2d v; v.x = re; v.y = im;
    *(v2d*)(out + idx) = v;                      // 16B aligned complex store
  }
}

__global__ void scatter_entries(const float* __restrict__ theta_x,
                                const float* __restrict__ theta_y,
                                double* __restrict__ out) {
  const int h = blockIdx.y;                      // 0..3 -> H1..H4
  const PairTab& T = g_tabs[h];
  const int i = blockIdx.x * blockDim.x + threadIdx.x;

  if (i < T.n) {
    const int r = T.r[i], c = T.c[i];
    put_entry(out, h, r, c, -1.0, 0.0);          // Jc = -J = -1
    put_entry(out, h, c, r, -1.0, 0.0);          // conj(Jc) = -1
  }

  if (i < NTW) {
    const double th = (double)((h & 1) ? *theta_x : *theta_y);
    const double cr = -cos(th);                  // Re(-e^{i th})
    const double si =  sin(th);
    put_entry(out, h, T.tr[i], T.tc[i], cr, -si);   // ph
    put_entry(out, h, T.tc[i], T.tr[i], cr,  si);   // conj(ph)
  }
}

// ---------------------------------------------------------------------------
// Harness entry point. d_in[0]=theta_x (f32 scalar), d_in[1]=theta_y.
// d_out = [4,4096,4096] complex128 viewed as doubles.
// ---------------------------------------------------------------------------
extern "C" void kernel_launch(void* const* d_in, const int* in_sizes, int n_in,
                              void* d_out, int out_size, void* d_ws, size_t ws_size,
                              hipStream_t stream) {
  (void)in_sizes; (void)n_in; (void)out_size; (void)d_ws; (void)ws_size;

  const float* theta_x = (const float*)d_in[0];
  const float* theta_y = (const float*)d_in[1];
  double* out = (double*)d_out;

  // Bandwidth-bound streaming memset: exact tiling, fully unrolled NT stores.
  fill_zero_nt<<<FILL_BLOCKS, FILL_THREADS, 0, stream>>>((v4u*)out);

  // Sparse scatter: one thread per pair, 4 blocks in y (one per Hamiltonian).
  dim3 grid((MAXP + 255) / 256, 4, 1);
  scatter_entries<<<grid, 256, 0, stream>>>(theta_x, theta_y, out);
}